// RLLoss_28458453303485
// MI455X (gfx1250) — compile-verified
//
#include <hip/hip_runtime.h>
#include <stdint.h>

#define GAMMA_ 0.985f
#define LAM_   0.95f
#define ENT_W_ 0.001f
#define B_ 1024
#define T_ 256
#define A_ 1024
#define BT_ (B_ * T_)

#define ROWS_PER_WAVE 4
#define NBLK_B (BT_ / (8 * ROWS_PER_WAVE))  // 8192
#define NBLK_A (B_ / 8)                     // 128 (one wave per batch row)

typedef __attribute__((ext_vector_type(2))) float v2f;
typedef __attribute__((ext_vector_type(4))) float f4;
typedef __attribute__((ext_vector_type(8))) float v8f;

// Wave-wide (32 lane) float sum using V_WMMA_F32_16X16X4_F32 with an all-ones
// B matrix.  D[m][n] = sum_k A[m][k]; a full column-sum of D equals the sum of
// ALL 64 A-slots (2 per lane), independent of the exact lane->(m,k) map.
// Lane n holds D[0..7][n] in c[0..7], lane n+16 holds D[8..15][n]: per-lane
// sum of c + shfl_xor(16) gives every lane the full 32-lane total.
// Requires EXEC == all ones (call sites are straight-line, full waves).
__device__ __forceinline__ float wave_sum_wmma(float x) {
  v2f a; a[0] = x;    a[1] = 0.0f;
  v2f b; b[0] = 1.0f; b[1] = 1.0f;
  v8f c = {};
  c = __builtin_amdgcn_wmma_f32_16x16x4_f32(false, a, false, b, (short)0, c,
                                            false, false);
  float s = ((c[0] + c[1]) + (c[2] + c[3])) + ((c[4] + c[5]) + (c[6] + c[7]));
  s += __shfl_xor(s, 16, 32);
  return s;
}

__device__ __forceinline__ float wave_max(float x) {
#pragma unroll
  for (int m = 16; m >= 1; m >>= 1) x = fmaxf(x, __shfl_xor(x, m, 32));
  return x;
}

// ---------------------------------------------------------------------------
// Kernel A: lambda-return scan as a parallel affine-recurrence scan.
// One WAVE per batch row b; lane L owns timesteps [8L, 8L+8).
//   ret_t = r'_t + a_t * ret_{t+1},  a_t = g*(1-d)*lam,
//   r'_t  = clip(r_t) + g*(1-d)*(1-lam)*v_{t+1}
// Each lane composes its chunk into one affine map (A,R); a 5-step cross-lane
// suffix-composition scan produces the carry ret_{8(L+1)}; lanes then replay
// their 8 local steps.  adv[b,t] = ret - v[b,t];  value-loss partial = adv^2.
// ---------------------------------------------------------------------------
__global__ void __launch_bounds__(256)
lambda_adv_kernel(const float* __restrict__ rewards,
                  const float* __restrict__ values,
                  const uint8_t* __restrict__ dones,
                  float* __restrict__ adv,
                  float* __restrict__ partA) {
  const int lane = threadIdx.x & 31;
  const int wv   = threadIdx.x >> 5;
  const int b    = blockIdx.x * 8 + wv;  // 0..B_-1 exactly

  const float*   rrow = rewards + (size_t)b * T_ + lane * 8;
  const float*   vrow = values + (size_t)b * (T_ + 1) + lane * 8;
  const uint8_t* drow = dones + (size_t)b * T_ + lane * 8;

  // coalesced chunk loads
  f4 rv0 = *(const f4*)(rrow);      // rewards 16B-aligned (row = 1024B)
  f4 rv1 = *(const f4*)(rrow + 4);
  const uint32_t* dp = (const uint32_t*)drow;  // 8B-aligned
  const uint32_t du0 = dp[0], du1 = dp[1];
  float v[9];
#pragma unroll
  for (int k = 0; k < 9; ++k) v[k] = vrow[k];  // vrow stride 257 -> scalar
  const float vT = values[(size_t)b * (T_ + 1) + T_];

  // per-step coefficients a_k (gain) and rp_k (bias), k local 0..7
  float ak[8], rp[8];
#pragma unroll
  for (int k = 0; k < 8; ++k) {
    float r  = (k < 4) ? rv0[k & 3] : rv1[k & 3];
    r = fminf(fmaxf(r, -10.0f), 10.0f);
    uint32_t byte = ((k < 4) ? du0 : du1) >> (8 * (k & 3));
    float nd = (byte & 0xFFu) ? 0.0f : 1.0f;  // 1-done
    ak[k] = GAMMA_ * LAM_ * nd;
    rp[k] = r + GAMMA_ * (1.0f - LAM_) * nd * v[k + 1];
  }

  // compose chunk: (A,R) maps ret_{chunk_end} -> ret_{chunk_start}
  float Ac = 1.0f, Rc = 0.0f;
#pragma unroll
  for (int k = 7; k >= 0; --k) {
    Rc = fmaf(ak[k], Rc, rp[k]);
    Ac = ak[k] * Ac;
  }

  // inclusive suffix-composition scan across lanes (identity padding)
#pragma unroll
  for (int d = 1; d < 32; d <<= 1) {
    float Ad = __shfl(Ac, lane + d, 32);
    float Rd = __shfl(Rc, lane + d, 32);
    bool valid = (lane + d) < 32;
    float Rn = fmaf(Ac, Rd, Rc);  // self (low t) after neighbor (high t)
    float An = Ac * Ad;
    Rc = valid ? Rn : Rc;
    Ac = valid ? An : Ac;
  }
  // exclusive carry from lane+1 = ret at own chunk end (lane 31: v_T)
  float cA = __shfl(Ac, lane + 1, 32);
  float cR = __shfl(Rc, lane + 1, 32);
  cA = (lane < 31) ? cA : 1.0f;
  cR = (lane < 31) ? cR : 0.0f;
  float ret = fmaf(cA, vT, cR);  // ret_{8(L+1)}

  // replay local steps, emit adv, accumulate value-loss partial
  float av[8];
  float vl = 0.0f;
#pragma unroll
  for (int k = 7; k >= 0; --k) {
    ret = fmaf(ak[k], ret, rp[k]);
    float a = ret - v[k];
    av[k] = a;
    vl = fmaf(a, a, vl);
  }
  f4 lo, hi;
  lo[0] = av[0]; lo[1] = av[1]; lo[2] = av[2]; lo[3] = av[3];
  hi[0] = av[4]; hi[1] = av[5]; hi[2] = av[6]; hi[3] = av[7];
  f4* ap = (f4*)(adv + (size_t)b * T_ + lane * 8);  // 16B-aligned
  ap[0] = lo;
  ap[1] = hi;

  float wsum = wave_sum_wmma(vl);  // full wave, EXEC all ones
  __shared__ float lds[8];
  if (lane == 0) lds[wv] = wsum;
  __syncthreads();
  if (threadIdx.x == 0) {
    float s = 0.0f;
    for (int i = 0; i < 8; ++i) s += lds[i];
    partA[blockIdx.x] = s;
  }
}

// ---------------------------------------------------------------------------
// Kernel B: one wave per 4 rows of A_=1024 logits (registers reused per row).
// Row lives in 8 float4 registers per lane (logits read exactly once,
// non-temporal -> don't thrash the 192MB L2 with a 1GB stream).  max via
// shuffles; sum(e^x), sum(x e^x) via WMMA wave reduction; action logit pulled
// from the already-loaded registers via lane shuffle.
// ---------------------------------------------------------------------------
__global__ void __launch_bounds__(256)
softmax_kernel(const float* __restrict__ logits,
               const int* __restrict__ actions,
               const float* __restrict__ adv,
               float* __restrict__ partB) {
  const int lane = threadIdx.x & 31;
  const int wv   = threadIdx.x >> 5;
  const long long row0 = (long long)blockIdx.x * (8 * ROWS_PER_WAVE) +
                         wv * ROWS_PER_WAVE;

  float s0 = 0.0f, s1 = 0.0f;  // wave-uniform accumulators
  for (int i = 0; i < ROWS_PER_WAVE; ++i) {
    const long long row = row0 + i;  // < BT_ exactly
    const f4* rp = (const f4*)(logits + (size_t)row * A_);
    f4 r[8];
#pragma unroll
    for (int j = 0; j < 8; ++j)
      r[j] = __builtin_nontemporal_load(rp + j * 32 + lane);

    float m = -3.402823466e+38f;
#pragma unroll
    for (int j = 0; j < 8; ++j)
      m = fmaxf(m, fmaxf(fmaxf(r[j][0], r[j][1]), fmaxf(r[j][2], r[j][3])));
    m = wave_max(m);

    const int act = actions[row];
    const int ja = act >> 7;         // element e=128j+4*lane+c
    const int la = (act >> 2) & 31;
    const int ca = act & 3;

    float cand = 0.0f, se = 0.0f, sxe = 0.0f;
#pragma unroll
    for (int j = 0; j < 8; ++j) {
      float x0 = r[j][0], x1 = r[j][1], x2 = r[j][2], x3 = r[j][3];
      float e0 = __expf(x0 - m), e1 = __expf(x1 - m);
      float e2 = __expf(x2 - m), e3 = __expf(x3 - m);
      se  += (e0 + e1) + (e2 + e3);
      sxe += (e0 * x0 + e1 * x1) + (e2 * x2 + e3 * x3);
      float cj = (ca == 0) ? x0 : (ca == 1) ? x1 : (ca == 2) ? x2 : x3;
      cand = (j == ja) ? cj : cand;
    }
    se  = wave_sum_wmma(se);
    sxe = wave_sum_wmma(sxe);

    const float logZ = m + __logf(se);
    const float ent  = logZ - sxe / se;              // -sum p*logp
    const float alp  = __shfl(cand, la, 32) - logZ;  // action log-prob
    s0 += alp * adv[row];
    s1 += ent;
  }

  __shared__ float lds[16];
  if (lane == 0) { lds[wv * 2] = s0; lds[wv * 2 + 1] = s1; }
  __syncthreads();
  if (threadIdx.x == 0) {
    float a0 = 0.0f, a1 = 0.0f;
    for (int i = 0; i < 8; ++i) { a0 += lds[i * 2]; a1 += lds[i * 2 + 1]; }
    partB[(size_t)blockIdx.x * 2]     = a0;
    partB[(size_t)blockIdx.x * 2 + 1] = a1;
  }
}

// ---------------------------------------------------------------------------
// Kernel C: deterministic final reduction in double precision -> scalar loss.
// ---------------------------------------------------------------------------
__global__ void __launch_bounds__(256)
final_kernel(const float* __restrict__ partB, int nB,
             const float* __restrict__ partA, int nA,
             float* __restrict__ out) {
  double s0 = 0.0, s1 = 0.0;
  for (int k = threadIdx.x; k < nB; k += 256) {
    s0 += (double)partB[2 * k];
    s1 += (double)partB[2 * k + 1];
  }
  __shared__ double l0[256], l1[256];
  l0[threadIdx.x] = s0;
  l1[threadIdx.x] = s1;
  __syncthreads();
  for (int off = 128; off > 0; off >>= 1) {
    if (threadIdx.x < off) {
      l0[threadIdx.x] += l0[threadIdx.x + off];
      l1[threadIdx.x] += l1[threadIdx.x + off];
    }
    __syncthreads();
  }
  if (threadIdx.x == 0) {
    double va = 0.0;
    for (int i = 0; i < nA; ++i) va += (double)partA[i];
    const double inv = 1.0 / (double)BT_;
    // loss = -mean(alp*adv) - ENT_W*mean(ent) + mean((v-lr)^2)
    double loss = -(l0[0] * inv) - (double)ENT_W_ * (l1[0] * inv) + va * inv;
    out[0] = (float)loss;
  }
}

extern "C" void kernel_launch(void* const* d_in, const int* in_sizes, int n_in,
                              void* d_out, int out_size, void* d_ws,
                              size_t ws_size, hipStream_t stream) {
  const float*   rewards = (const float*)d_in[0];
  const float*   values  = (const float*)d_in[1];
  const float*   logits  = (const float*)d_in[2];
  const uint8_t* dones   = (const uint8_t*)d_in[3];  // jnp.bool_ = 1 byte
  const int*     actions = (const int*)d_in[4];
  float* out = (float*)d_out;

  // workspace layout (floats): adv[BT_], partB[2*NBLK_B], partA[NBLK_A] ~1.1MB
  float* adv   = (float*)d_ws;
  float* partB = adv + BT_;
  float* partA = partB + 2 * (size_t)NBLK_B;

  lambda_adv_kernel<<<NBLK_A, 256, 0, stream>>>(rewards, values, dones, adv,
                                                partA);
  softmax_kernel<<<NBLK_B, 256, 0, stream>>>(logits, actions, adv, partB);
  final_kernel<<<1, 256, 0, stream>>>(partB, NBLK_B, partA, NBLK_A, out);
}